// EdgeVariantGF_30021821399295
// MI455X (gfx1250) — compile-verified
//
#include <hip/hip_runtime.h>

// ---------------------------------------------------------------------------
// EdgeVariantGF hybrid filter for MI455X (gfx1250, wave32, WMMA).
//   B=64, F=16, E=1, K=4, G=16, N=256, M_SEL=128
// Strategy:
//   * EVGF chain  : per-(f,g) block, v in LDS [batch][node] (batch-major ->
//                   B-operands are single conflict-free b64 LDS loads with
//                   immediate offsets, D publishes merge to ds_store_b128),
//                   fp32 WMMA V_WMMA_F32_16X16X4_F32 along the 256-deep K dim.
//                   weightEV (256 MB, streamed once -> HBM-bound) is staged
//                   into wave-private LDS slices with coalesced
//                   GLOBAL_LOAD_ASYNC_TO_LDS_B128 (+ s_wait_asynccnt),
//                   double-buffered, nontemporal.
//   * LSI chain   : per-g block, Z = x * S^k via the same WMMA shape
//                   (S is 256 KB, L2-resident -> direct loads are fine).
//   * mask (sp * hybrid) precomputed once into d_ws (L2-resident, reused).
//   * y initialized to bias, partial sums combined with global f32 atomics.
// ---------------------------------------------------------------------------

typedef float v2f __attribute__((ext_vector_type(2)));
typedef float v8f __attribute__((ext_vector_type(8)));

#define Bq 64
#define Fq 16
#define Kq 4
#define Gq 16
#define Nq 256
#define MSEL 128
#define ZTOL 1e-9f

#define NSTR 260         // LDS node stride for v[batch][node] / Z[batch][node]
                         //   (256 nodes + 4 pad: (b*4 + m) mod 64 -> no conflicts)
#define ZSTR 260
#define WSTR 36          // LDS row stride (floats) for staged weight slices:
                         //   l16*36 + c (c in 0..3) mod 64 hits all 64 banks
#define WROWS 32         // rows per wave band
#define WBUF (WROWS * WSTR)            // floats per staging buffer (1152)
#define WOFF (2 * Bq * NSTR)           // float offset of staging area (33280)

static __device__ inline v8f wmma_f32_4(v2f a, v2f b, v8f c) {
  // D(16x16,f32) = A(16x4,f32) x B(4x16,f32) + C
  return __builtin_amdgcn_wmma_f32_16x16x4_f32(
      /*neg_a=*/false, a, /*neg_b=*/false, b,
      /*c_mod=*/(short)0, c, /*reuse_a=*/false, /*reuse_b=*/false);
}

// Issue one 4 KB slice (32 rows x 32 m-values) of Phi_k rows [rowBase, +32)
// into this wave's private LDS staging buffer with 8 coalesced async B128 ops.
static __device__ inline void stage_slice_async(const float* __restrict__ Wk,
                                                unsigned wregionBytes,
                                                int rowBase, int s, int lane,
                                                int bsel) {
  const int rsub  = lane >> 3;   // 0..3 : row within group of 4
  const int chunk = lane & 7;    // 0..7 : 16-byte chunk within 128-byte row seg
  const unsigned lbase = wregionBytes + (unsigned)(bsel * (WBUF * 4));
#pragma unroll
  for (int i = 0; i < 8; ++i) {
    const int rl = i * 4 + rsub;                                   // 0..31
    const unsigned voff =
        (unsigned)(((rowBase + rl) * Nq + s * 32) * 4 + chunk * 16);
    const unsigned laddr = lbase + (unsigned)(rl * (WSTR * 4) + chunk * 16);
    asm volatile("global_load_async_to_lds_b128 %0, %1, %2 th:TH_LOAD_NT"
                 :: "v"(laddr), "v"(voff), "s"(Wk)
                 : "memory");
  }
}

static __device__ inline void wait_async(int pending) {
  if (pending)
    asm volatile("s_wait_asynccnt 0x8" ::: "memory");  // oldest 8 (slice s) done
  else
    asm volatile("s_wait_asynccnt 0x0" ::: "memory");
}

// ---------------------------------------------------------------------------
// mask[n*N+m] = ((|S[n,m]| + I) > tol) * hybrid(n,m)
// ---------------------------------------------------------------------------
__global__ __launch_bounds__(256)
void mask_kernel(const float* __restrict__ S, float* __restrict__ mask) {
  int i = blockIdx.x * 256 + threadIdx.x;       // i < N*N
  int n = i >> 8, m = i & 255;
  float v = fabsf(S[i]) + ((n == m) ? 1.0f : 0.0f);
  float hm = ((n < MSEL) | (m < MSEL)) ? 1.0f : 0.0f;
  mask[i] = ((v > ZTOL) ? 1.0f : 0.0f) * hm;
}

// ---------------------------------------------------------------------------
// y[b,f,n] = bias[f]
// ---------------------------------------------------------------------------
__global__ __launch_bounds__(256)
void init_y_kernel(const float* __restrict__ bias, float* __restrict__ y) {
  int i = blockIdx.x * 256 + threadIdx.x;       // i < B*F*N
  int f = (i >> 8) & (Fq - 1);
  y[i] = bias[f];
}

// ---------------------------------------------------------------------------
// EVGF: one block per (f,g).  acc = v0 + Phi1 v0 + Phi2 Phi1 v0 + ...
// LDS: 2*64*260 floats (v double-buffer) + 8 waves * 2 * 1152 floats (staging)
//      = 206,848 bytes (< 320 KB WGP LDS)
// ---------------------------------------------------------------------------
__global__ __launch_bounds__(256)
void evgf_kernel(const float* __restrict__ x, const float* __restrict__ wEV,
                 const float* __restrict__ mask, float* __restrict__ y) {
  extern __shared__ float lds[];
  float* buf0 = lds;                 // [Bq][NSTR]
  float* buf1 = lds + Bq * NSTR;

  const int f = blockIdx.x / Gq;
  const int g = blockIdx.x % Gq;
  const int tid  = threadIdx.x;
  const int wave = tid >> 5;
  const int lane = tid & 31;
  const int half = lane >> 4;        // 0: K 0/1, 1: K 2/3 (A layout)
  const int l16  = lane & 15;

  const int rowBase = wave * WROWS;  // this wave's 32-row band of Phi_k
  float* wslice0 = lds + WOFF + wave * 2 * WBUF;
  const unsigned wregionBytes = (unsigned)(size_t)(const void*)wslice0;

  // ---- tap 0: v0[b][n] = Wdiag(f,g,n) * x[b,g,n] for n < M_SEL ----
  {
    const float* W0 = wEV + ((size_t)(f * Kq + 0) * Gq + g) * (size_t)(Nq * Nq);
    const int n = tid;
    float w = 0.0f;
    if (n < MSEL) w = __builtin_nontemporal_load(&W0[(size_t)n * Nq + n]);
    for (int b = 0; b < Bq; ++b) {
      float val = (n < MSEL) ? w * x[((size_t)b * Gq + g) * Nq + n] : 0.0f;
      buf0[b * NSTR + n] = val;
    }
  }
  __syncthreads();

  // wave owns row tiles {2w, 2w+1} x batch tiles {0..3}
  const int rt0 = wave * 2;
  v8f acc[2][4];
  for (int r = 0; r < 2; ++r)
    for (int bt = 0; bt < 4; ++bt)
      for (int j = 0; j < 8; ++j) {
        int n = (rt0 + r) * 16 + j + half * 8;   // D layout: row j(+8), col l16
        int b = bt * 16 + l16;
        acc[r][bt][j] = buf0[b * NSTR + n];
      }

  float* vcur  = buf0;
  float* vnext = buf1;

  for (int k = 1; k < Kq; ++k) {
    const float* Wk = wEV + ((size_t)(f * Kq + k) * Gq + g) * (size_t)(Nq * Nq);
    v8f d[2][4];
    for (int r = 0; r < 2; ++r)
      for (int bt = 0; bt < 4; ++bt)
        d[r][bt] = (v8f){0.f, 0.f, 0.f, 0.f, 0.f, 0.f, 0.f, 0.f};

    int bsel = 0;
    stage_slice_async(Wk, wregionBytes, rowBase, 0, lane, 0);

    for (int s = 0; s < 8; ++s) {                 // 8 slices of 32 m-values
      if (s < 7) stage_slice_async(Wk, wregionBytes, rowBase, s + 1, lane,
                                   bsel ^ 1);
      wait_async(s < 7);                          // slice s resident in LDS
      const int mbase = s * 32;

      // Per-slice base pointers: every inner-loop access becomes a
      // constant-offset load off a single base register.
      const v2f* wp = (const v2f*)(wslice0 + bsel * WBUF + l16 * WSTR +
                                   half * 2);
      const v2f* mp = (const v2f*)(mask + (size_t)(rowBase + l16) * Nq +
                                   mbase + half * 2);
      const v2f* vp = (const v2f*)(vcur + l16 * NSTR + mbase + half * 2);

#pragma unroll
      for (int msub = 0; msub < 32; msub += 4) {
        const int h = msub >> 1;                  // v2f element index
        // A tiles: staged weights (LDS) * mask (global, L2), pk_mul
        v2f a[2];
        a[0] = wp[h]                  * mp[h];
        a[1] = wp[(16 * WSTR) / 2 + h] * mp[(16 * Nq) / 2 + h];
        // B tiles: batched b64 LDS loads (immediate offsets), then WMMA burst
        v2f bb[4];
        bb[0] = vp[h];
        bb[1] = vp[(16 * NSTR) / 2 + h];
        bb[2] = vp[(32 * NSTR) / 2 + h];
        bb[3] = vp[(48 * NSTR) / 2 + h];
#pragma unroll
        for (int bt = 0; bt < 4; ++bt) {
          d[0][bt] = wmma_f32_4(a[0], bb[bt], d[0][bt]);
          d[1][bt] = wmma_f32_4(a[1], bb[bt], d[1][bt]);
        }
      }
      bsel ^= 1;
    }

    // acc += v_k ; publish v_k (8 consecutive floats/lane -> ds_store_b128)
    for (int r = 0; r < 2; ++r)
      for (int bt = 0; bt < 4; ++bt) {
        acc[r][bt] += d[r][bt];
        const int nbase = (rt0 + r) * 16 + half * 8;
        const int b = bt * 16 + l16;
        for (int j = 0; j < 8; ++j)
          vnext[b * NSTR + nbase + j] = d[r][bt][j];
      }
    __syncthreads();
    float* t = vcur; vcur = vnext; vnext = t;
  }

  // y[b,f,n] += acc  (sum over g via atomics)
  for (int r = 0; r < 2; ++r)
    for (int bt = 0; bt < 4; ++bt)
      for (int j = 0; j < 8; ++j) {
        int n = (rt0 + r) * 16 + j + half * 8;
        int b = bt * 16 + l16;
        atomicAdd(&y[((size_t)b * Fq + f) * Nq + n], acc[r][bt][j]);
      }
}

// ---------------------------------------------------------------------------
// LSI: one block per g.  Z_k = Z_{k-1} @ S ;  y[b,f,n] += wLSI[f,k,g]*Z_k[b,n]
// LDS: double-buffered Z[batch][node] (2 * 64 * 260 * 4B = 133,120 B)
// ---------------------------------------------------------------------------
static __device__ inline void lsi_emit(const float* __restrict__ Z,
                                       const float* __restrict__ wLSI,
                                       float* __restrict__ y,
                                       int g, int k, int tid) {
  float wl[Fq];
  for (int f = 0; f < Fq; ++f) wl[f] = wLSI[(f * Kq + k) * Gq + g];
  for (int it = 0; it < (Bq * Nq) / 256; ++it) {
    int idx = it * 256 + tid;
    int b = idx >> 8, n = idx & 255;
    float val = Z[b * ZSTR + n];
    float* yp = &y[((size_t)b * Fq) * Nq + n];
    for (int f = 0; f < Fq; ++f) atomicAdd(yp + (size_t)f * Nq, wl[f] * val);
  }
}

__global__ __launch_bounds__(256)
void lsi_kernel(const float* __restrict__ x, const float* __restrict__ wLSI,
                const float* __restrict__ S, float* __restrict__ y) {
  extern __shared__ float lds[];
  float* bufA = lds;                 // [Bq][ZSTR]
  float* bufB = lds + Bq * ZSTR;

  const int g = blockIdx.x;
  const int tid  = threadIdx.x;
  const int wave = tid >> 5;
  const int lane = tid & 31;
  const int half = lane >> 4;
  const int l16  = lane & 15;

  // Z0 = x[:, g, :]
  for (int it = 0; it < (Bq * Nq) / 256; ++it) {
    int idx = it * 256 + tid;
    int b = idx >> 8, m = idx & 255;
    bufA[b * ZSTR + m] = x[((size_t)b * Gq + g) * Nq + m];
  }
  __syncthreads();
  lsi_emit(bufA, wLSI, y, g, 0, tid);

  // wave owns batch tile (wave>>1) and 8 node tiles starting at (wave&1)*8
  const int bt = wave >> 1;
  const int ntBase = (wave & 1) * 8;
  const int brow = bt * 16 + l16;

  float* Zcur  = bufA;
  float* Znext = bufB;
  for (int k = 1; k < Kq; ++k) {
    v8f d[8];
    for (int t = 0; t < 8; ++t)
      d[t] = (v8f){0.f, 0.f, 0.f, 0.f, 0.f, 0.f, 0.f, 0.f};

    const v2f* ap = (const v2f*)(Zcur + brow * ZSTR + half * 2);

    for (int m0 = 0; m0 < Nq; m0 += 4) {
      const int mc = m0 + half * 2;
      // A tile: single b64 LDS load (constant offset off one base)
      v2f a = ap[m0 >> 1];
      // B tiles: batch all 8 S fetches (L2-resident), then WMMA burst
      const float* s0 = S + (size_t)mc * Nq + l16;
      v2f bb[8];
#pragma unroll
      for (int t = 0; t < 8; ++t) {
        const int n0 = (ntBase + t) * 16;
        bb[t].x = s0[n0];
        bb[t].y = s0[Nq + n0];
      }
#pragma unroll
      for (int t = 0; t < 8; ++t) d[t] = wmma_f32_4(a, bb[t], d[t]);
    }

    for (int t = 0; t < 8; ++t) {
      const int n0 = (ntBase + t) * 16;
      for (int j = 0; j < 8; ++j) {
        int b = bt * 16 + j + half * 8;
        Znext[b * ZSTR + n0 + l16] = d[t][j];
      }
    }
    __syncthreads();
    float* t = Zcur; Zcur = Znext; Znext = t;
    lsi_emit(Zcur, wLSI, y, g, k, tid);
  }
}

// ---------------------------------------------------------------------------
extern "C" void kernel_launch(void* const* d_in, const int* in_sizes, int n_in,
                              void* d_out, int out_size, void* d_ws, size_t ws_size,
                              hipStream_t stream) {
  (void)in_sizes; (void)n_in; (void)out_size; (void)ws_size;
  const float* x    = (const float*)d_in[0];   // (B,G,N)
  const float* wEV  = (const float*)d_in[1];   // (F,E,K,G,N,N)
  const float* wLSI = (const float*)d_in[2];   // (F,E,K,G)
  const float* bias = (const float*)d_in[3];   // (F,1)
  const float* S    = (const float*)d_in[4];   // (E,N,N)
  float* y    = (float*)d_out;                 // (B,F,N)
  float* mask = (float*)d_ws;                  // N*N floats of scratch

  mask_kernel<<<(Nq * Nq) / 256, 256, 0, stream>>>(S, mask);
  init_y_kernel<<<(Bq * Fq * Nq) / 256, 256, 0, stream>>>(bias, y);

  const size_t ldsEV =
      (size_t)(2 * Bq * NSTR + 8 * 2 * WBUF) * sizeof(float);  // 206,848 B
  evgf_kernel<<<Fq * Gq, 256, ldsEV, stream>>>(x, wEV, mask, y);

  const size_t ldsLSI = (size_t)2 * Bq * ZSTR * sizeof(float); // 133,120 B
  lsi_kernel<<<Gq, 256, ldsLSI, stream>>>(x, wLSI, S, y);
}